// MatchingLayer_54614804136321
// MI455X (gfx1250) — compile-verified
//
#include <hip/hip_runtime.h>
#include <hip/hip_bf16.h>

// MatchingLayer (BiMPM) for MI455X / gfx1250.
// fp32 WMMA (v_wmma_f32_16x16x4_f32) for the three matmul stages; VALU for the
// max-plus stage; wave32 shuffle reductions for the per-row mpm features.

#define BB 16
#define SS 128   // SP == SH
#define HH 200
#define PP 20
#define FF 105
#define EPSF 1e-8f

typedef float v2f __attribute__((ext_vector_type(2)));
typedef float v8f __attribute__((ext_vector_type(8)));

__device__ __forceinline__ float wave_sum(float v) {
#pragma unroll
  for (int o = 16; o > 0; o >>= 1) v += __shfl_xor(v, o, 32);
  return v;
}

// ---------------------------------------------------------------------------
// prep: dst = src * mask (zero padded rows), row L2 norms, w1-weighted norms
// grid = B*S blocks of 32 (one wave per row)
// ---------------------------------------------------------------------------
__global__ void prep_kernel(const float* __restrict__ src, const int* __restrict__ mask,
                            const float* __restrict__ w1, float* __restrict__ dst,
                            float* __restrict__ nrm, float* __restrict__ wn) {
  __shared__ float row[HH];
  int bs = blockIdx.x;
  int lane = threadIdx.x;
  float m = (float)mask[bs];
  const float* srow = src + (size_t)bs * HH;
  float* drow = dst + (size_t)bs * HH;
  float acc = 0.f;
  for (int h = lane; h < HH; h += 32) {
    float v = srow[h] * m;
    drow[h] = v; row[h] = v; acc += v * v;
  }
  acc = wave_sum(acc);
  if (lane == 0) nrm[bs] = sqrtf(acc);
  __syncthreads();
  for (int p = 0; p < PP; p++) {
    float a = 0.f;
    for (int h = lane; h < HH; h += 32) {
      float ww = w1[p * HH + h]; float v = row[h];
      a += ww * ww * v * v;
    }
    a = wave_sum(a);
    if (lane == 0) wn[bs * PP + p] = sqrtf(a);
  }
}

// ---------------------------------------------------------------------------
// gather last valid token rows: cpl[b]=cp[b,len_p-1], chl[b]=ch[b,len_h-1]
// grid = B blocks of 64
// ---------------------------------------------------------------------------
__global__ void gather_last_kernel(const float* __restrict__ cp, const float* __restrict__ ch,
                                   const int* __restrict__ mask_p, const int* __restrict__ mask_h,
                                   float* __restrict__ cpl, float* __restrict__ chl) {
  int b = blockIdx.x;
  int lp = 0, lh = 0;
  for (int i = 0; i < SS; i++) { lp += mask_p[b * SS + i]; lh += mask_h[b * SS + i]; }
  int ip = (lp > 0) ? lp - 1 : 0;
  int ih = (lh > 0) ? lh - 1 : 0;
  for (int h = threadIdx.x; h < HH; h += blockDim.x) {
    cpl[b * HH + h] = cp[((size_t)b * SS + ip) * HH + h];
    chl[b * HH + h] = ch[((size_t)b * SS + ih) * HH + h];
  }
}

// ---------------------------------------------------------------------------
// cos[b,x,y] = (cp_x . ch_y) / max(|cp_x||ch_y|, eps)  via f32 WMMA 16x16x4
// grid = B*8*8 blocks of 32 (one wave per 16x16 tile)
// ---------------------------------------------------------------------------
__global__ void cos_wmma_kernel(const float* __restrict__ cp, const float* __restrict__ ch,
                                const float* __restrict__ np, const float* __restrict__ nh,
                                float* __restrict__ cosO) {
  int wid = blockIdx.x;
  int b = wid >> 6;
  int x0 = ((wid >> 3) & 7) << 4;
  int y0 = (wid & 7) << 4;
  int lane = threadIdx.x, half = lane >> 4, l16 = lane & 15;
  const float* arow = cp + ((size_t)(b * SS) + x0 + l16) * HH + 2 * half;
  const float* brow = ch + ((size_t)(b * SS) + y0 + l16) * HH + 2 * half;
  v8f acc = {0, 0, 0, 0, 0, 0, 0, 0};
  for (int k = 0; k < HH; k += 4) {
    v2f a = *(const v2f*)(arow + k);
    v2f bv = *(const v2f*)(brow + k);
    acc = __builtin_amdgcn_wmma_f32_16x16x4_f32(false, a, false, bv, (short)0, acc, false, false);
  }
  float nhv = nh[b * SS + y0 + l16];
#pragma unroll
  for (int r = 0; r < 8; r++) {
    int m = r + 8 * half;
    float d = fmaxf(np[b * SS + x0 + m] * nhv, EPSF);
    cosO[((size_t)(b * SS) + x0 + m) * SS + y0 + l16] = acc[r] / d;
  }
}

// ---------------------------------------------------------------------------
// cos row/col reductions -> features 0,1 and cossum
// grid = B*S blocks of 128
// ---------------------------------------------------------------------------
__global__ void cos_reduce_kernel(const float* __restrict__ cosM, const int* __restrict__ mask,
                                  float* __restrict__ cossum, float* __restrict__ out,
                                  int transposed, int side) {
  int b = blockIdx.x / SS, r = blockIdx.x % SS;
  int t = threadIdx.x;
  float v = transposed ? cosM[((size_t)b * SS + t) * SS + r]
                       : cosM[((size_t)b * SS + r) * SS + t];
  __shared__ float smax[128], ssum[128];
  smax[t] = v; ssum[t] = v;
  __syncthreads();
  for (int o = 64; o > 0; o >>= 1) {
    if (t < o) { smax[t] = fmaxf(smax[t], smax[t + o]); ssum[t] += ssum[t + o]; }
    __syncthreads();
  }
  if (t == 0) {
    float m = (float)mask[b * SS + r];
    cossum[b * SS + r] = ssum[0];
    float* o2 = out + ((size_t)(side * BB + b) * SS + r) * FF;
    o2[0] = m * smax[0];
    o2[1] = m * (ssum[0] / (float)SS);
  }
}

// ---------------------------------------------------------------------------
// att_mean = (cos [or cos^T] @ V) / max(cossum, eps)  via f32 WMMA, K=128
// grid = B*8*13 blocks of 32
// ---------------------------------------------------------------------------
__global__ void attmean_wmma_kernel(const float* __restrict__ cosM, const float* __restrict__ V,
                                    const float* __restrict__ csum, float* __restrict__ attO,
                                    int aStrideM, int aStrideK) {
  int wid = blockIdx.x;
  int b = wid / (8 * 13);
  int rem = wid % (8 * 13);
  int m0 = (rem / 13) * 16;
  int n0 = (rem % 13) * 16;
  int lane = threadIdx.x, half = lane >> 4, l16 = lane & 15;
  const float* cb = cosM + (size_t)b * SS * SS;
  int nIdx = n0 + l16;
  bool nOK = nIdx < HH;
  v8f acc = {0, 0, 0, 0, 0, 0, 0, 0};
  for (int k = 0; k < SS; k += 4) {
    int ka = k + 2 * half;
    v2f a, bv;
    a.x = cb[(size_t)(m0 + l16) * aStrideM + (size_t)(ka + 0) * aStrideK];
    a.y = cb[(size_t)(m0 + l16) * aStrideM + (size_t)(ka + 1) * aStrideK];
    bv.x = nOK ? V[((size_t)b * SS + ka + 0) * HH + nIdx] : 0.f;
    bv.y = nOK ? V[((size_t)b * SS + ka + 1) * HH + nIdx] : 0.f;
    acc = __builtin_amdgcn_wmma_f32_16x16x4_f32(false, a, false, bv, (short)0, acc, false, false);
  }
  if (nOK) {
#pragma unroll
    for (int r = 0; r < 8; r++) {
      int m = m0 + r + 8 * half;
      float d = fmaxf(csum[b * SS + m], EPSF);
      attO[((size_t)b * SS + m) * HH + nIdx] = acc[r] / d;
    }
  }
}

// ---------------------------------------------------------------------------
// att_max[b,r,h] = mask[b,r] * max_k ( cos[...] * V[b,k,h] )   (max-plus)
// grid = B*S blocks of 256 (h < 200)
// ---------------------------------------------------------------------------
__global__ void attmax_kernel(const float* __restrict__ cosM, const float* __restrict__ V,
                              const int* __restrict__ mask, float* __restrict__ attO,
                              int cRowStride, int cColStride) {
  int b = blockIdx.x / SS, r = blockIdx.x % SS;
  int h = threadIdx.x;
  if (h >= HH) return;
  const float* cb = cosM + (size_t)b * SS * SS;
  float mx = -3.0e38f;
  for (int k = 0; k < SS; k++) {
    float c = cb[(size_t)r * cRowStride + (size_t)k * cColStride];
    float v = V[((size_t)b * SS + k) * HH + h];
    mx = fmaxf(mx, c * v);
  }
  float m = (float)mask[b * SS + r];
  attO[((size_t)b * SS + r) * HH + h] = m * mx;
}

// ---------------------------------------------------------------------------
// mpm features: mv_1 (cosine, per-norm clamp) + 20 weighted cosines
// grid = B*S blocks of 32 (one wave per row)
// ---------------------------------------------------------------------------
__global__ void mpm_kernel(const float* __restrict__ v1, const float* __restrict__ v2,
                           const float* __restrict__ w, int v2_bstride, int v2_sstride,
                           float* __restrict__ out, int side, int foff) {
  __shared__ float r1[HH], r2[HH];
  int bs = blockIdx.x;
  int b = bs / SS, s = bs % SS;
  int lane = threadIdx.x;
  const float* p1 = v1 + (size_t)bs * HH;
  const float* p2 = v2 + (size_t)b * v2_bstride + (size_t)s * v2_sstride;
  float d11 = 0, d22 = 0, d12 = 0;
  for (int h = lane; h < HH; h += 32) {
    float a = p1[h], c = p2[h];
    r1[h] = a; r2[h] = c;
    d11 += a * a; d22 += c * c; d12 += a * c;
  }
  d11 = wave_sum(d11); d22 = wave_sum(d22); d12 = wave_sum(d12);
  __syncthreads();
  float* orow = out + ((size_t)(side * BB + b) * SS + s) * FF + foff;
  if (lane == 0)
    orow[0] = d12 / (fmaxf(sqrtf(d11), EPSF) * fmaxf(sqrtf(d22), EPSF));
  for (int p = 0; p < PP; p++) {
    float s11 = 0, s22 = 0, s12 = 0;
    for (int h = lane; h < HH; h += 32) {
      float ww = w[p * HH + h]; float w2 = ww * ww;
      float a = r1[h], c = r2[h];
      s11 += w2 * a * a; s22 += w2 * c * c; s12 += w2 * a * c;
    }
    s11 = wave_sum(s11); s22 = wave_sum(s22); s12 = wave_sum(s12);
    if (lane == 0)
      orow[1 + p] = s12 / (fmaxf(sqrtf(s11), EPSF) * fmaxf(sqrtf(s22), EPSF));
  }
}

// ---------------------------------------------------------------------------
// maxpool match: for fixed (b,p,x-tile), loop y-tiles; WMMA the weighted dot,
// normalize, fuse max+mean over y in registers. grid = B*P*8 blocks of 32.
// ---------------------------------------------------------------------------
__global__ void maxpool_wmma_kernel(const float* __restrict__ v1, const float* __restrict__ v2,
                                    const float* __restrict__ w,
                                    const float* __restrict__ wnA, const float* __restrict__ wnB,
                                    const int* __restrict__ mask, float* __restrict__ out,
                                    int side) {
  int wid = blockIdx.x;
  int b = wid / (PP * 8);
  int p = (wid / 8) % PP;
  int x0 = (wid % 8) * 16;
  int lane = threadIdx.x, half = lane >> 4, l16 = lane & 15;
  float rmax[8], rsum[8], wA[8];
#pragma unroll
  for (int r = 0; r < 8; r++) {
    rmax[r] = -3.0e38f; rsum[r] = 0.f;
    wA[r] = wnA[((size_t)b * SS + x0 + r + 8 * half) * PP + p];
  }
  const float* a_base = v1 + ((size_t)b * SS + x0 + l16) * HH;
  const float* wp = w + p * HH;
  for (int y0 = 0; y0 < SS; y0 += 16) {
    const float* b_base = v2 + ((size_t)b * SS + y0 + l16) * HH;
    v8f acc = {0, 0, 0, 0, 0, 0, 0, 0};
    for (int k = 0; k < HH; k += 4) {
      int ka = k + 2 * half;
      float w0v = wp[ka], w1v = wp[ka + 1];
      v2f a, bv;
      a.x = w0v * w0v * a_base[ka];
      a.y = w1v * w1v * a_base[ka + 1];
      bv = *(const v2f*)(b_base + ka);
      acc = __builtin_amdgcn_wmma_f32_16x16x4_f32(false, a, false, bv, (short)0, acc, false, false);
    }
    float wB = wnB[((size_t)b * SS + y0 + l16) * PP + p];
#pragma unroll
    for (int r = 0; r < 8; r++) {
      float mv = acc[r] / fmaxf(wA[r] * wB, EPSF);
      rmax[r] = fmaxf(rmax[r], mv);
      rsum[r] += mv;
    }
  }
#pragma unroll
  for (int r = 0; r < 8; r++) {
    for (int o = 8; o > 0; o >>= 1) {
      rmax[r] = fmaxf(rmax[r], __shfl_xor(rmax[r], o, 32));
      rsum[r] += __shfl_xor(rsum[r], o, 32);
    }
  }
  if (l16 == 0) {
#pragma unroll
    for (int r = 0; r < 8; r++) {
      int x = x0 + r + 8 * half;
      float m = (float)mask[b * SS + x];
      float* orow = out + ((size_t)(side * BB + b) * SS + x) * FF;
      orow[23 + p] = m * rmax[r];
      orow[43 + p] = m * (rsum[r] / (float)SS);
    }
  }
}

// ---------------------------------------------------------------------------
extern "C" void kernel_launch(void* const* d_in, const int* in_sizes, int n_in,
                              void* d_out, int out_size, void* d_ws, size_t ws_size,
                              hipStream_t stream) {
  const float* ctx_p = (const float*)d_in[0];
  const float* ctx_h = (const float*)d_in[1];
  const float* w0 = (const float*)d_in[2];
  const float* w1 = (const float*)d_in[3];
  const float* w2 = (const float*)d_in[4];
  const float* w3 = (const float*)d_in[5];
  const int* mask_p = (const int*)d_in[6];
  const int* mask_h = (const int*)d_in[7];
  float* out = (float*)d_out;
  float* ws = (float*)d_ws;

  const size_t SZ_ROWS = (size_t)BB * SS * HH;   // 409600
  float* cp = ws;                 ws += SZ_ROWS;
  float* ch = ws;                 ws += SZ_ROWS;
  float* n_p = ws;                ws += BB * SS;
  float* n_h = ws;                ws += BB * SS;
  float* wn1_p = ws;              ws += (size_t)BB * SS * PP;
  float* wn1_h = ws;              ws += (size_t)BB * SS * PP;
  float* cpl = ws;                ws += BB * HH;
  float* chl = ws;                ws += BB * HH;
  float* cosM = ws;               ws += (size_t)BB * SS * SS;
  float* cossum_y = ws;           ws += BB * SS;
  float* cossum_x = ws;           ws += BB * SS;
  float* att_mean_h = ws;         ws += SZ_ROWS;
  float* att_mean_p = ws;         ws += SZ_ROWS;
  float* att_max_h = ws;          ws += SZ_ROWS;
  float* att_max_p = ws;          ws += SZ_ROWS;

  // 1. prep both sides (mask multiply, norms, w1-weighted norms)
  prep_kernel<<<BB * SS, 32, 0, stream>>>(ctx_p, mask_p, w1, cp, n_p, wn1_p);
  prep_kernel<<<BB * SS, 32, 0, stream>>>(ctx_h, mask_h, w1, ch, n_h, wn1_h);

  // 2. gather last valid rows
  gather_last_kernel<<<BB, 64, 0, stream>>>(cp, ch, mask_p, mask_h, cpl, chl);

  // 3. pairwise cosine via WMMA
  cos_wmma_kernel<<<BB * 64, 32, 0, stream>>>(cp, ch, n_p, n_h, cosM);

  // 4. cos reductions (features 0,1 + cossums)
  cos_reduce_kernel<<<BB * SS, 128, 0, stream>>>(cosM, mask_p, cossum_y, out, 0, 0);
  cos_reduce_kernel<<<BB * SS, 128, 0, stream>>>(cosM, mask_h, cossum_x, out, 1, 1);

  // 5. attentive means via WMMA (cos @ ch, cos^T @ cp)
  attmean_wmma_kernel<<<BB * 8 * 13, 32, 0, stream>>>(cosM, ch, cossum_y, att_mean_h, SS, 1);
  attmean_wmma_kernel<<<BB * 8 * 13, 32, 0, stream>>>(cosM, cp, cossum_x, att_mean_p, 1, SS);

  // 6. max-attentive vectors (max-plus)
  attmax_kernel<<<BB * SS, 256, 0, stream>>>(cosM, ch, mask_p, att_max_h, SS, 1);
  attmax_kernel<<<BB * SS, 256, 0, stream>>>(cosM, cp, mask_h, att_max_p, 1, SS);

  // 7. mpm feature groups: full (w0, foff 2), attentive (w2, foff 63),
  //    max-attentive (w3, foff 84); each for premise (side 0) and hyp (side 1)
  mpm_kernel<<<BB * SS, 32, 0, stream>>>(cp, chl, w0, HH, 0, out, 0, 2);
  mpm_kernel<<<BB * SS, 32, 0, stream>>>(ch, cpl, w0, HH, 0, out, 1, 2);
  mpm_kernel<<<BB * SS, 32, 0, stream>>>(cp, att_mean_h, w2, SS * HH, HH, out, 0, 63);
  mpm_kernel<<<BB * SS, 32, 0, stream>>>(ch, att_mean_p, w2, SS * HH, HH, out, 1, 63);
  mpm_kernel<<<BB * SS, 32, 0, stream>>>(cp, att_max_h, w3, SS * HH, HH, out, 0, 84);
  mpm_kernel<<<BB * SS, 32, 0, stream>>>(ch, att_max_p, w3, SS * HH, HH, out, 1, 84);

  // 8. maxpool match via WMMA with fused max/mean reductions (features 23..62)
  maxpool_wmma_kernel<<<BB * PP * 8, 32, 0, stream>>>(cp, ch, w1, wn1_p, wn1_h, mask_p, out, 0);
  maxpool_wmma_kernel<<<BB * PP * 8, 32, 0, stream>>>(ch, cp, w1, wn1_h, wn1_p, mask_h, out, 1);
}